// LightGCN_41326175322814
// MI455X (gfx1250) — compile-verified
//
#include <hip/hip_runtime.h>
#include <hip/hip_bf16.h>
#include <cstdint>

#define LG_D 128
#define LG_LAYERS 3

typedef __attribute__((ext_vector_type(2))) float v2f;
typedef __attribute__((ext_vector_type(4))) float v4f;
typedef __attribute__((ext_vector_type(8))) float v8f;

__device__ __forceinline__ int lg_bcast_i(int x, int l) {
  return __builtin_amdgcn_readlane(x, l);
}
__device__ __forceinline__ float lg_bcast_f(float x, int l) {
  return __int_as_float(__builtin_amdgcn_readlane(__float_as_int(x), l));
}

// ---- CDNA5 async-to-LDS path detection (device pass only; host pass and
// toolchains without the builtins fall back to the synchronous path) ----
#if __has_builtin(__builtin_amdgcn_global_load_async_to_lds_b32)
#define LG_ASYNC 1
#if __has_builtin(__builtin_amdgcn_s_wait_asynccnt)
#define LG_WAIT_ASYNC() __builtin_amdgcn_s_wait_asynccnt(0)
#else
#define LG_WAIT_ASYNC() asm volatile("s_wait_asynccnt 0x0" ::: "memory")
#endif
#else
#define LG_ASYNC 0
#endif

// Builtin signature (from round-2 diagnostics): both pointer params are int*,
// arg0 = address_space(1) global source, arg1 = address_space(3) LDS dest,
// then two integer immediates (offset, cpol).
#define LG_AS1I(p) ((__attribute__((address_space(1))) int*)(uintptr_t)(p))
#define LG_AS3I(p) ((__attribute__((address_space(3))) int*)(void*)(p))

// ---- elementwise helpers (float4 vectorized, grid-stride) ----

__global__ __launch_bounds__(256) void lgcn_init(const v4f* __restrict__ x,
                                                 v4f* __restrict__ a,
                                                 v4f* __restrict__ acc, long n4) {
  long i = (long)blockIdx.x * blockDim.x + threadIdx.x;
  const long s = (long)gridDim.x * blockDim.x;
  for (; i < n4; i += s) { v4f v = x[i]; a[i] = v; acc[i] = v; }
}

__global__ __launch_bounds__(256) void lgcn_zero(v4f* __restrict__ p, long n4) {
  long i = (long)blockIdx.x * blockDim.x + threadIdx.x;
  const long s = (long)gridDim.x * blockDim.x;
  v4f z = {0.f, 0.f, 0.f, 0.f};
  for (; i < n4; i += s) p[i] = z;
}

__global__ __launch_bounds__(256) void lgcn_add(v4f* __restrict__ acc,
                                                const v4f* __restrict__ b, long n4) {
  long i = (long)blockIdx.x * blockDim.x + threadIdx.x;
  const long s = (long)gridDim.x * blockDim.x;
  for (; i < n4; i += s) {
    v4f v = acc[i], w = b[i];
    v.x += w.x; v.y += w.y; v.z += w.z; v.w += w.w;
    acc[i] = v;
  }
}

// ---- SpMM push: one wave handles batches of 32 edges; per edge, the wave
// gathers one 512B embedding row (global_load_b128 per lane, L2-resident
// since the 51.2MB table fits the 192MB L2) and scatters it with native f32
// atomics. Edge metadata is double-buffered into per-wave LDS with
// global_load_async_to_lds_b32 (ASYNCcnt) when available, overlapping the
// HBM edge stream with compute; otherwise it is loaded coalesced and
// broadcast with v_readlane. A 2-deep gather pipeline keeps b128 loads in
// flight during the atomic scatter. ----

__global__ __launch_bounds__(256) void lgcn_spmm(const int* __restrict__ erow,
                                                 const int* __restrict__ ecol,
                                                 const float* __restrict__ eval,
                                                 const float* __restrict__ embin,
                                                 float* __restrict__ out, int E) {
  const int lane = threadIdx.x & 31;
  const long gwave = ((long)blockIdx.x * blockDim.x + threadIdx.x) >> 5;
  const long nwave = ((long)gridDim.x * blockDim.x) >> 5;
  const long stride = nwave * 32;
  const long base0 = gwave * 32;

#if LG_ASYNC
  __shared__ int sh_row[8][2][32];
  __shared__ int sh_col[8][2][32];
  __shared__ float sh_val[8][2][32];
  const int wv = threadIdx.x >> 5;

  auto issue = [&](long b, int which) {
    const long e = b + lane;
    if (e < E) {
      __builtin_amdgcn_global_load_async_to_lds_b32(LG_AS1I(erow + e),
                                                    LG_AS3I(&sh_row[wv][which][lane]), 0, 0);
      __builtin_amdgcn_global_load_async_to_lds_b32(LG_AS1I(ecol + e),
                                                    LG_AS3I(&sh_col[wv][which][lane]), 0, 0);
      __builtin_amdgcn_global_load_async_to_lds_b32(LG_AS1I(eval + e),
                                                    LG_AS3I(&sh_val[wv][which][lane]), 0, 0);
    }
  };

  int buf = 0;
  if (base0 < E) issue(base0, 0);

  for (long base = base0; base < E; base += stride) {
    LG_WAIT_ASYNC();
    asm volatile("" ::: "memory");  // keep LDS reads after the wait

    const long nxt = base + stride;
    if (nxt < E) issue(nxt, buf ^ 1);

    // prefetch the batch after next into L2 (global_prefetch_b8)
    if (nxt + stride < E) {
      __builtin_prefetch(&erow[nxt + stride + lane], 0, 1);
      __builtin_prefetch(&ecol[nxt + stride + lane], 0, 1);
      __builtin_prefetch(&eval[nxt + stride + lane], 0, 1);
    }

    const long rem = (long)E - base;
    const int cnt = rem < 32 ? (int)rem : 32;
    const int* crow = sh_row[wv][buf];
    const int* ccol = sh_col[wv][buf];
    const float* cval = sh_val[wv][buf];

    // 2-deep gather pipeline
    v4f g0 = *(const v4f*)(embin + (size_t)ccol[0] * LG_D + lane * 4);
    v4f g1 = g0;
    if (cnt > 1) g1 = *(const v4f*)(embin + (size_t)ccol[1] * LG_D + lane * 4);

    for (int j = 0; j < cnt; ++j) {
      const v4f gc = g0;
      g0 = g1;
      if (j + 2 < cnt)
        g1 = *(const v4f*)(embin + (size_t)ccol[j + 2] * LG_D + lane * 4);
      const int rj = crow[j];
      const float vj = cval[j];
      float* dst = out + (size_t)rj * LG_D + lane * 4;
      unsafeAtomicAdd(dst + 0, vj * gc.x);
      unsafeAtomicAdd(dst + 1, vj * gc.y);
      unsafeAtomicAdd(dst + 2, vj * gc.z);
      unsafeAtomicAdd(dst + 3, vj * gc.w);
    }
    buf ^= 1;
  }
#else
  for (long base = base0; base < E; base += stride) {
    const long e = base + lane;
    int r = 0, c = 0;
    float v = 0.f;
    if (e < E) { r = erow[e]; c = ecol[e]; v = eval[e]; }

    if (base + stride < E) {
      __builtin_prefetch(&erow[base + stride + lane], 0, 1);
      __builtin_prefetch(&ecol[base + stride + lane], 0, 1);
      __builtin_prefetch(&eval[base + stride + lane], 0, 1);
    }

    const long rem = (long)E - base;
    const int cnt = rem < 32 ? (int)rem : 32;

    v4f g0 = *(const v4f*)(embin + (size_t)lg_bcast_i(c, 0) * LG_D + lane * 4);
    v4f g1 = g0;
    if (cnt > 1) g1 = *(const v4f*)(embin + (size_t)lg_bcast_i(c, 1) * LG_D + lane * 4);

    for (int j = 0; j < cnt; ++j) {
      const v4f gc = g0;
      g0 = g1;
      if (j + 2 < cnt)
        g1 = *(const v4f*)(embin + (size_t)lg_bcast_i(c, j + 2) * LG_D + lane * 4);
      const int rj = lg_bcast_i(r, j);
      const float vj = lg_bcast_f(v, j);
      float* dst = out + (size_t)rj * LG_D + lane * 4;
      unsafeAtomicAdd(dst + 0, vj * gc.x);
      unsafeAtomicAdd(dst + 1, vj * gc.y);
      unsafeAtomicAdd(dst + 2, vj * gc.z);
      unsafeAtomicAdd(dst + 3, vj * gc.w);
    }
  }
#endif
}

// ---- Finalize: acc *= 0.25 on the matrix pipe. One V_WMMA_F32_16X16X4_F32
// per 16 elements: A[m][k] = acc[base+m] for all k (replicated across K, so
// any K-ordering of the A layout is fine), B[k][j] = 1/16 everywhere (any B
// row order is fine). D[m][j] = 4 * acc[base+m] / 16 = 0.25*acc[base+m],
// exact (power-of-two scale, all cross terms exactly zero). Only the
// documented C/D 16x16 layout is relied on for the store. ----

__global__ __launch_bounds__(256) void lgcn_finalize_wmma(float* __restrict__ acc,
                                                          long nd) {
  const int lane = threadIdx.x & 31;
  const long gwave = ((long)blockIdx.x * blockDim.x + threadIdx.x) >> 5;
  const long nwave = ((long)gridDim.x * blockDim.x) >> 5;
  const long ntiles = nd >> 4;  // nd = N*128, divisible by 16
  const int m = lane & 15;

  v2f bmat;
  bmat[0] = 0.0625f;
  bmat[1] = 0.0625f;

  for (long t = gwave; t < ntiles; t += nwave) {
    const long base = t << 4;
    const float a = acc[base + m];
    v2f amat;
    amat[0] = a;
    amat[1] = a;
    v8f cmat = {};
    cmat = __builtin_amdgcn_wmma_f32_16x16x4_f32(
        /*neg_a=*/false, amat, /*neg_b=*/false, bmat,
        /*c_mod=*/(short)0, cmat, /*reuse_a=*/false, /*reuse_b=*/false);
    // D layout: VGPR r, lanes 0-15 -> M=r, lanes 16-31 -> M=r+8 (all N equal).
    if (m < 8) {
      const int r = lane & 7;
      float v = cmat[0];
      if (r == 1) v = cmat[1];
      if (r == 2) v = cmat[2];
      if (r == 3) v = cmat[3];
      if (r == 4) v = cmat[4];
      if (r == 5) v = cmat[5];
      if (r == 6) v = cmat[6];
      if (r == 7) v = cmat[7];
      acc[base + r + ((lane < 16) ? 0 : 8)] = v;
    }
  }
}

// ---- host-side orchestration ----

extern "C" void kernel_launch(void* const* d_in, const int* in_sizes, int n_in,
                              void* d_out, int out_size, void* d_ws, size_t ws_size,
                              hipStream_t stream) {
  (void)n_in; (void)out_size; (void)ws_size;

  const float* x = (const float*)d_in[0];
  const int* erow = (const int*)d_in[1];
  const int* ecol = (const int*)d_in[2];
  const float* eval = (const float*)d_in[3];

  const long nd = (long)in_sizes[0];  // N * 128
  const int E = in_sizes[1];
  const long n4 = nd >> 2;

  float* bufA = (float*)d_ws;        // current layer embedding (N*D floats)
  float* bufB = bufA + nd;           // next layer embedding   (N*D floats)
  float* acc = (float*)d_out;        // running sum of layer embeddings

  const int tpb = 256;
  const int ew_blocks = 2048;
  const int spmm_blocks = 4096;

  lgcn_init<<<ew_blocks, tpb, 0, stream>>>((const v4f*)x, (v4f*)bufA, (v4f*)acc, n4);

  for (int layer = 0; layer < LG_LAYERS; ++layer) {
    lgcn_zero<<<ew_blocks, tpb, 0, stream>>>((v4f*)bufB, n4);
    lgcn_spmm<<<spmm_blocks, tpb, 0, stream>>>(erow, ecol, eval, bufA, bufB, E);
    lgcn_add<<<ew_blocks, tpb, 0, stream>>>((v4f*)acc, (const v4f*)bufB, n4);
    float* t = bufA; bufA = bufB; bufB = t;
  }

  lgcn_finalize_wmma<<<ew_blocks, tpb, 0, stream>>>(acc, nd);
}